// CLNGCN_7430293422686
// MI455X (gfx1250) — compile-verified
//
#include <hip/hip_runtime.h>
#include <hip/hip_bf16.h>
#include <math.h>

typedef __attribute__((ext_vector_type(2))) float v2f;
typedef __attribute__((ext_vector_type(8))) float v8f;

__device__ __forceinline__ float gelu_exact(float x) {
    // torch exact GELU: 0.5*x*(1+erf(x/sqrt(2)))
    return 0.5f * x * (1.0f + erff(x * 0.70710678118654752440f));
}

// One wave32 does the whole graph.
// WMMA mapping (V_WMMA_F32_16X16X4_F32, full f32 precision):
//   layer1: D[16x16] = A(16x4, x in row 0) x B(4x16, W1^T tile) ; K=8 -> 2 chunks, N=32 -> 2 tiles
//   layer2: K=32 -> 8 accumulating WMMAs, N tile holds the 8 outputs in cols 0..7
// A 16x4 f32 layout: lanes 0-15: M=lane, VGPR0=K0, VGPR1=K1 ; lanes 16-31: M=lane-16, VGPR0=K2, VGPR1=K3
// B 4x16 f32 layout (C/D striping rule): VGPR0: K0 (lanes 0-15) / K2 (lanes 16-31); VGPR1: K1 / K3
// D 16x16 f32 layout: VGPR0 lanes 0-15 = row M=0, N=lane
__global__ __launch_bounds__(32)
void clngcn_wmma_kernel(const float* __restrict__ cli,
                        const float* __restrict__ c1w, const float* __restrict__ c1b,
                        const float* __restrict__ c2w, const float* __restrict__ c2b,
                        const float* __restrict__ m1w1, const float* __restrict__ m1b1,
                        const float* __restrict__ m1w2, const float* __restrict__ m1b2,
                        const float* __restrict__ m2w1, const float* __restrict__ m2b1,
                        const float* __restrict__ m2w2, const float* __restrict__ m2b2,
                        const float* __restrict__ g1w, const float* __restrict__ g1b,
                        const float* __restrict__ g2w, const float* __restrict__ g2b,
                        float* __restrict__ out)
{
    __shared__ float xs[8];     // cli
    __shared__ float gs1[32];   // mlp1 hidden (pre/post GELU)
    __shared__ float gs2[32];   // mlp2 hidden
    __shared__ float ss[8];     // cli_ss (mlp1 output)
    __shared__ float mm[8];     // cli_mm (mlp2 output)

    const int  lane = (int)threadIdx.x;      // 0..31, single wave32
    const bool hi   = (lane >= 16);
    const int  nl   = lane & 15;

    if (lane < 8) xs[lane] = cli[lane];
    __syncthreads();

    // ---- fragment builders ----
    // A tile with only row M=0 populated (vector x / hidden act), K chunk at kbase
    auto afrag = [&](const float* src, int kbase) -> v2f {
        v2f a; a.x = 0.f; a.y = 0.f;
        if (lane == 0)  { a.x = src[kbase + 0]; a.y = src[kbase + 1]; }
        if (lane == 16) { a.x = src[kbase + 2]; a.y = src[kbase + 3]; }
        return a;
    };
    // layer1 B: W1 is [32,8] row-major, y[n] = sum_k x[k]*W1[n*8+k]; tile t covers n in [16t,16t+16)
    auto bfrag_l1 = [&](const float* w1, int t, int kbase) -> v2f {
        v2f b;
        const int n  = 16 * t + nl;
        const int k0 = kbase + (hi ? 2 : 0);
        b.x = w1[n * 8 + k0];
        b.y = w1[n * 8 + k0 + 1];
        return b;
    };
    // layer2 B: W2 is [8,32] row-major, y[j] = sum_n h[n]*W2[j*32+n]; cols 8..15 zero
    auto bfrag_l2 = [&](const float* w2, int kbase) -> v2f {
        v2f b; b.x = 0.f; b.y = 0.f;
        const int k0 = kbase + (hi ? 2 : 0);
        if (nl < 8) {
            b.x = w2[nl * 32 + k0];
            b.y = w2[nl * 32 + k0 + 1];
        }
        return b;
    };

    const v8f z = {0.f, 0.f, 0.f, 0.f, 0.f, 0.f, 0.f, 0.f};

    // ---------------- MLP layer 1 (both MLPs interleaved): 8 WMMAs ----------------
    v8f c10 = z, c11 = z, c20 = z, c21 = z;
    for (int kb = 0; kb < 8; kb += 4) {
        v2f a = afrag(xs, kb);
        v2f b;
        b   = bfrag_l1(m1w1, 0, kb);
        c10 = __builtin_amdgcn_wmma_f32_16x16x4_f32(false, a, false, b, (short)0, c10, false, false);
        b   = bfrag_l1(m1w1, 1, kb);
        c11 = __builtin_amdgcn_wmma_f32_16x16x4_f32(false, a, false, b, (short)0, c11, false, false);
        b   = bfrag_l1(m2w1, 0, kb);
        c20 = __builtin_amdgcn_wmma_f32_16x16x4_f32(false, a, false, b, (short)0, c20, false, false);
        b   = bfrag_l1(m2w1, 1, kb);
        c21 = __builtin_amdgcn_wmma_f32_16x16x4_f32(false, a, false, b, (short)0, c21, false, false);
    }
    // D row 0 lives in element 0 of lanes 0..15
    if (lane < 16) {
        gs1[nl]      = c10[0];
        gs1[16 + nl] = c11[0];
        gs2[nl]      = c20[0];
        gs2[16 + nl] = c21[0];
    }
    __syncthreads();

    // bias + exact GELU, all 32 lanes (one hidden unit each, per MLP)
    gs1[lane] = gelu_exact(gs1[lane] + m1b1[lane]);
    gs2[lane] = gelu_exact(gs2[lane] + m2b1[lane]);
    __syncthreads();

    // ---------------- MLP layer 2 (two independent 8-deep chains): 16 WMMAs ----------------
    v8f d1 = z, d2 = z;
    for (int kb = 0; kb < 32; kb += 4) {
        v2f a1 = afrag(gs1, kb);
        v2f a2 = afrag(gs2, kb);
        v2f b1 = bfrag_l2(m1w2, kb);
        v2f b2 = bfrag_l2(m2w2, kb);
        d1 = __builtin_amdgcn_wmma_f32_16x16x4_f32(false, a1, false, b1, (short)0, d1, false, false);
        d2 = __builtin_amdgcn_wmma_f32_16x16x4_f32(false, a2, false, b2, (short)0, d2, false, false);
    }
    if (lane < 8) {
        ss[lane] = d1[0] + m1b2[lane];   // cli_ss
        mm[lane] = d2[0] + m2b2[lane];   // cli_mm
    }
    __syncthreads();

    // ---------------- epilogue: each lane owns column j = lane & 7 ----------------
    const float w_c1 = c1w[0], b_c1 = c1b[0];
    const float w_c2 = c2w[0], b_c2 = c2b[0];
    const float w_g1 = g1w[0], b_g1 = g1b[0];
    const float w_g2 = g2w[0], b_g2 = g2b[0];
    const int j = lane & 7;

    float sigma[8], alpha[8], seg[8];
    float dot = 0.f;                     // dot(cli_ss, alpha)
    for (int i = 0; i < 8; ++i) {
        const float xi = xs[i];
        sigma[i] = xi * w_c1 + b_c1;
        alpha[i] = xi * w_c2 + b_c2;
        seg[i]   = fmaxf(xi * w_g1 + b_g1, 0.f);
        dot     += ss[i] * alpha[i];
    }

    // S[i] = similarity_c[i][j] + diag_att[i][j]
    //      = (mm[i]*dot)*alpha[j] + mm[i]*ss[j]*sigma[i]*sigma[j]
    float S[8];
    float mx = -__builtin_inff();
    for (int i = 0; i < 8; ++i) {
        const float sim_c = (mm[i] * dot) * alpha[j];
        const float datt  = (mm[i] * ss[j]) * (sigma[i] * sigma[j]);
        S[i] = sim_c + datt;
        mx = fmaxf(mx, S[i]);
    }

    // softmax over i (axis=1), fused with seg @ similarity
    float sum = 0.f, num = 0.f;
    for (int i = 0; i < 8; ++i) {
        const float e = expf(S[i] - mx);
        sum += e;
        num += seg[i] * e;
    }
    const float seg_sim = num / sum;
    const float o = fmaxf(seg_sim * w_g2 + b_g2, 0.f) + seg[j];

    if (lane < 8) out[lane] = o;
}

extern "C" void kernel_launch(void* const* d_in, const int* in_sizes, int n_in,
                              void* d_out, int out_size, void* d_ws, size_t ws_size,
                              hipStream_t stream) {
    (void)in_sizes; (void)n_in; (void)out_size; (void)d_ws; (void)ws_size;
    clngcn_wmma_kernel<<<1, 32, 0, stream>>>(
        (const float*)d_in[0],                            // cli
        (const float*)d_in[1],  (const float*)d_in[2],    // conv1 w,b
        (const float*)d_in[3],  (const float*)d_in[4],    // conv2 w,b
        (const float*)d_in[5],  (const float*)d_in[6],    // mlp1 w1,b1
        (const float*)d_in[7],  (const float*)d_in[8],    // mlp1 w2,b2
        (const float*)d_in[9],  (const float*)d_in[10],   // mlp2 w1,b1
        (const float*)d_in[11], (const float*)d_in[12],   // mlp2 w2,b2
        (const float*)d_in[13], (const float*)d_in[14],   // gcn1 w,b
        (const float*)d_in[15], (const float*)d_in[16],   // gcn2 w,b
        (float*)d_out);
}